// GraphAttentionNetwork_55078660604364
// MI455X (gfx1250) — compile-verified
//
#include <hip/hip_runtime.h>
#include <hip/hip_bf16.h>
#include <float.h>
#include <math.h>

#define HEADS 4
#define CPH 32
#define HC 128           // HEADS*CPH
#define NEG_SLOPE 0.2f
#define EPSV 1e-5f

typedef float v2f __attribute__((ext_vector_type(2)));
typedef float v8f __attribute__((ext_vector_type(8)));

// ---------- monotonic float <-> uint encoding for atomic max ----------
__device__ __forceinline__ unsigned enc_f32(float f) {
    unsigned u = __float_as_uint(f);
    return (u & 0x80000000u) ? ~u : (u | 0x80000000u);
}
__device__ __forceinline__ float dec_f32(unsigned k) {
    unsigned u = (k & 0x80000000u) ? (k ^ 0x80000000u) : ~k;
    return __uint_as_float(u);
}

__device__ __forceinline__ void edge_sd(long long e, const int* __restrict__ src,
                                        const int* __restrict__ dst, int E,
                                        int* s, int* d) {
    if (e < (long long)E) { *s = src[e]; *d = dst[e]; }
    else { int v = (int)(e - E); *s = v; *d = v; }
}

// ---------------------------------------------------------------------
// GEMM: Hout[M x 128] = A[M x 128] @ W[128 x 128]  (row-major, fp32 WMMA)
// One wave computes one 16x16 tile; block = 256 thr = 8 waves = 16x128 slab.
// A-frag 16x4:  lanes 0-15 M=0..15, v0=K0/K2 (lo/hi half-wave), v1=K1/K3
// B-frag 4x16:  v0 = row K0 (lanes 0-15) / K2 (lanes 16-31), v1 = K1/K3
// C/D 16x16:    vgpr r: lanes 0-15 -> M=r, lanes 16-31 -> M=r+8, N=lane%16
// ---------------------------------------------------------------------
__global__ __launch_bounds__(256) void k_gemm_wmma(
    const float* __restrict__ A, const float* __restrict__ W,
    float* __restrict__ Hout, int M)
{
    const int lane = threadIdx.x & 31;
    const int wave = threadIdx.x >> 5;   // 0..7 (uniform per wave)
    const int m0   = blockIdx.x * 16;
    const int n0   = wave * 16;
    const int hi   = lane >> 4;          // 0/1 half-wave
    const int l16  = lane & 15;
    const bool full = (m0 + 16 <= M);    // uniform across wave

    int row = m0 + l16;
    if (row >= M) row = M - 1;           // clamp loads only (tail blocks)
    const float* __restrict__ arow = A + (size_t)row * HC;

    v8f acc = {0.f, 0.f, 0.f, 0.f, 0.f, 0.f, 0.f, 0.f};

#pragma unroll 4
    for (int k = 0; k < HC; k += 4) {
        const int kk = k + 2 * hi;
        v2f av, bv;
        av.x = arow[kk];
        av.y = arow[kk + 1];
        bv.x = W[(size_t)kk * HC + n0 + l16];
        bv.y = W[(size_t)(kk + 1) * HC + n0 + l16];
        acc = __builtin_amdgcn_wmma_f32_16x16x4_f32(
            /*neg_a=*/false, av, /*neg_b=*/false, bv,
            /*c_mod=*/(short)0, acc, /*reuse_a=*/false, /*reuse_b=*/false);
    }

    float* __restrict__ outp = Hout + (size_t)(m0 + 8 * hi) * HC + n0 + l16;
    if (full) {
        // fast path: unguarded, 8 coalesced b32 stores
#pragma unroll
        for (int r = 0; r < 8; ++r) outp[(size_t)r * HC] = acc[r];
    } else {
#pragma unroll
        for (int r = 0; r < 8; ++r) {
            int ro = m0 + r + 8 * hi;
            if (ro < M) Hout[(size_t)ro * HC + n0 + l16] = acc[r];
        }
    }
}

// ---------------------------------------------------------------------
// Attention logits: al_s[n,h] = sum_c h[n,h,c]*a_s[h,c] ; same for a_d
// ---------------------------------------------------------------------
__global__ void k_attn_logits(const float* __restrict__ Hm,
                              const float* __restrict__ a_s,
                              const float* __restrict__ a_d,
                              float* __restrict__ als, float* __restrict__ ald,
                              int Nn)
{
    int idx = blockIdx.x * blockDim.x + threadIdx.x;
    if (idx >= Nn * HEADS) return;
    int n = idx >> 2, h = idx & 3;
    const float4* hp = (const float4*)(Hm + (size_t)n * HC + h * CPH);
    const float4* as = (const float4*)(a_s + h * CPH);
    const float4* ad = (const float4*)(a_d + h * CPH);
    float ss = 0.f, sd = 0.f;
#pragma unroll
    for (int c = 0; c < CPH / 4; ++c) {
        float4 v = hp[c], a = as[c], b = ad[c];
        ss += v.x * a.x + v.y * a.y + v.z * a.z + v.w * a.w;
        sd += v.x * b.x + v.y * b.y + v.z * b.z + v.w * b.w;
    }
    als[idx] = ss;
    ald[idx] = sd;
}

// ---------------------------------------------------------------------
// Init accumulators: agg=0 (vectorized), denom=0, mkey=0 (enc-space min)
// one thread per 4 floats of agg
// ---------------------------------------------------------------------
__global__ void k_init(float4* __restrict__ agg4, float* __restrict__ denom,
                       unsigned* __restrict__ mkey, int Nn)
{
    long long idx = (long long)blockIdx.x * blockDim.x + threadIdx.x;
    if (idx < (long long)Nn * (HC / 4)) agg4[idx] = make_float4(0.f, 0.f, 0.f, 0.f);
    if (idx < (long long)Nn * HEADS)    { denom[idx] = 0.f; mkey[idx] = 0u; }
}

// ---------------------------------------------------------------------
// Pass 1: per-(edge,head) leaky-relu logit -> atomic max per (dst,head)
// ---------------------------------------------------------------------
__global__ void k_edge_max(const int* __restrict__ src, const int* __restrict__ dst,
                           const float* __restrict__ als, const float* __restrict__ ald,
                           unsigned* __restrict__ mkey, int E, int Etot)
{
    long long idx = (long long)blockIdx.x * blockDim.x + threadIdx.x;
    if (idx >= (long long)Etot * HEADS) return;
    long long e = idx >> 2; int h = (int)(idx & 3);
    int s, d; edge_sd(e, src, dst, E, &s, &d);
    float ev = als[s * HEADS + h] + ald[d * HEADS + h];
    ev = (ev < 0.f) ? NEG_SLOPE * ev : ev;
    atomicMax(&mkey[d * HEADS + h], enc_f32(ev));
}

// ---------------------------------------------------------------------
// Pass 2: denom[dst,h] += exp(e - m[dst,h])
// ---------------------------------------------------------------------
__global__ void k_edge_expsum(const int* __restrict__ src, const int* __restrict__ dst,
                              const float* __restrict__ als, const float* __restrict__ ald,
                              const unsigned* __restrict__ mkey, float* __restrict__ denom,
                              int E, int Etot)
{
    long long idx = (long long)blockIdx.x * blockDim.x + threadIdx.x;
    if (idx >= (long long)Etot * HEADS) return;
    long long e = idx >> 2; int h = (int)(idx & 3);
    int s, d; edge_sd(e, src, dst, E, &s, &d);
    int dh = d * HEADS + h;
    float ev = als[s * HEADS + h] + ald[dh];
    ev = (ev < 0.f) ? NEG_SLOPE * ev : ev;
    atomicAdd(&denom[dh], expf(ev - dec_f32(mkey[dh])));
}

// ---------------------------------------------------------------------
// Pass 3: agg[dst,k] += h[src,k] * alpha(edge, head(k))
// One thread per (edge, 4 channels): b128 gather of h[src], 4 atomics.
// ---------------------------------------------------------------------
__global__ void k_edge_msg(const int* __restrict__ src, const int* __restrict__ dst,
                           const float* __restrict__ als, const float* __restrict__ ald,
                           const unsigned* __restrict__ mkey, const float* __restrict__ denom,
                           const float* __restrict__ Hm, float* __restrict__ agg,
                           int E, int Etot)
{
    long long idx = (long long)blockIdx.x * blockDim.x + threadIdx.x;
    if (idx >= (long long)Etot * (HC / 4)) return;
    long long e = idx >> 5;              // 32 threads per edge
    int q  = (int)(idx & 31);            // quad index 0..31
    int k4 = q * 4;                      // channel base
    int h  = k4 >> 5;                    // head 0..3
    int s, d; edge_sd(e, src, dst, E, &s, &d);
    int dh = d * HEADS + h;
    float ev = als[s * HEADS + h] + ald[dh];
    ev = (ev < 0.f) ? NEG_SLOPE * ev : ev;
    float alpha = expf(ev - dec_f32(mkey[dh])) / denom[dh];

    float4 v = *(const float4*)(Hm + (size_t)s * HC + k4);
    float* ap = agg + (size_t)d * HC + k4;
    atomicAdd(ap + 0, v.x * alpha);
    atomicAdd(ap + 1, v.y * alpha);
    atomicAdd(ap + 2, v.z * alpha);
    atomicAdd(ap + 3, v.w * alpha);
}

// ---------------------------------------------------------------------
// Epilogue: out = bias-add, then optionally LayerNorm + ReLU (in place ok)
// One 128-thread block per node; LDS tree reduce for mean / E[x^2].
// ---------------------------------------------------------------------
__global__ __launch_bounds__(HC) void k_epilogue(
    const float* __restrict__ agg, const float* __restrict__ bias,
    const float* __restrict__ g, const float* __restrict__ bln,
    float* __restrict__ out, int apply_ln)
{
    __shared__ float s1[HC];
    __shared__ float s2[HC];
    const int n = blockIdx.x;
    const int t = threadIdx.x;
    float v = agg[(size_t)n * HC + t] + bias[t];
    if (!apply_ln) {                 // uniform branch across block
        out[(size_t)n * HC + t] = v;
        return;
    }
    s1[t] = v;
    s2[t] = v * v;
    __syncthreads();
#pragma unroll
    for (int off = HC / 2; off > 0; off >>= 1) {
        if (t < off) { s1[t] += s1[t + off]; s2[t] += s2[t + off]; }
        __syncthreads();
    }
    float mu  = s1[0] * (1.f / HC);
    float var = s2[0] * (1.f / HC) - mu * mu;
    float y = (v - mu) * rsqrtf(var + EPSV) * g[t] + bln[t];
    out[(size_t)n * HC + t] = fmaxf(y, 0.f);
}

// ---------------------------------------------------------------------
extern "C" void kernel_launch(void* const* d_in, const int* in_sizes, int n_in,
                              void* d_out, int out_size, void* d_ws, size_t ws_size,
                              hipStream_t stream)
{
    const float* x    = (const float*)d_in[0];
    const int*   esrc = (const int*)d_in[1];
    const int*   edst = (const int*)d_in[2];
    const float* W[3]  = {(const float*)d_in[3],  (const float*)d_in[7],  (const float*)d_in[11]};
    const float* As[3] = {(const float*)d_in[4],  (const float*)d_in[8],  (const float*)d_in[12]};
    const float* Ad[3] = {(const float*)d_in[5],  (const float*)d_in[9],  (const float*)d_in[13]};
    const float* Bi[3] = {(const float*)d_in[6],  (const float*)d_in[10], (const float*)d_in[14]};
    const float* LnG[2] = {(const float*)d_in[15], (const float*)d_in[17]};
    const float* LnB[2] = {(const float*)d_in[16], (const float*)d_in[18]};

    const int Nn   = in_sizes[0] / HC;
    const int E    = in_sizes[1];
    const int Etot = E + Nn;

    // workspace layout (floats)
    float*    hbuf  = (float*)d_ws;                       // Nn*HC
    float*    agg   = hbuf + (size_t)Nn * HC;             // Nn*HC
    float*    als   = agg  + (size_t)Nn * HC;             // Nn*HEADS
    float*    ald   = als  + (size_t)Nn * HEADS;          // Nn*HEADS
    unsigned* mkey  = (unsigned*)(ald + (size_t)Nn * HEADS); // Nn*HEADS
    float*    denom = (float*)(mkey + (size_t)Nn * HEADS);   // Nn*HEADS

    const int TB = 256;
    const unsigned gGemm = (unsigned)((Nn + 15) / 16);
    const unsigned gAttn = (unsigned)(((long long)Nn * HEADS + TB - 1) / TB);
    const unsigned gInit = (unsigned)(((long long)Nn * (HC / 4) + TB - 1) / TB);
    const unsigned gEH   = (unsigned)(((long long)Etot * HEADS + TB - 1) / TB);
    const unsigned gMsg  = (unsigned)(((long long)Etot * (HC / 4) + TB - 1) / TB);

    const float* cur = x;
    for (int l = 0; l < 3; ++l) {
        float* layer_out = (l == 2) ? (float*)d_out : agg;

        k_gemm_wmma<<<gGemm, 256, 0, stream>>>(cur, W[l], hbuf, Nn);
        k_attn_logits<<<gAttn, TB, 0, stream>>>(hbuf, As[l], Ad[l], als, ald, Nn);
        k_init<<<gInit, TB, 0, stream>>>((float4*)layer_out, denom, mkey, Nn);
        k_edge_max<<<gEH, TB, 0, stream>>>(esrc, edst, als, ald, mkey, E, Etot);
        k_edge_expsum<<<gEH, TB, 0, stream>>>(esrc, edst, als, ald, mkey, denom, E, Etot);
        k_edge_msg<<<gMsg, TB, 0, stream>>>(esrc, edst, als, ald, mkey, denom,
                                            hbuf, layer_out, E, Etot);
        if (l < 2) {
            k_epilogue<<<(unsigned)Nn, HC, 0, stream>>>(layer_out, Bi[l],
                                                        LnG[l], LnB[l],
                                                        layer_out, 1);
            cur = layer_out;   // agg feeds the next layer's GEMM
        } else {
            k_epilogue<<<(unsigned)Nn, HC, 0, stream>>>(layer_out, Bi[l],
                                                        LnG[0], LnB[0],
                                                        layer_out, 0);
        }
    }
}